// Attn_60224031425225
// MI455X (gfx1250) — compile-verified
//
#include <hip/hip_runtime.h>
#include <hip/hip_bf16.h>

#define N_    1024
#define M_    1024
#define ENC_  512
#define ATTN_ 256

typedef __attribute__((ext_vector_type(2))) float v2f;
typedef __attribute__((ext_vector_type(8))) float v8f;

typedef __attribute__((address_space(1))) int g_i32;   // global (printed __device__)
typedef __attribute__((address_space(3))) int s_i32;   // LDS

#if __has_builtin(__builtin_amdgcn_global_load_async_to_lds_b32) && \
    __has_builtin(__builtin_amdgcn_s_wait_asynccnt)
#define HAS_ASYNC_LDS 1
#else
#define HAS_ASYNC_LDS 0
#endif

__device__ __forceinline__ float fast_tanh(float x) {
#if __has_builtin(__builtin_amdgcn_tanhf)
    return __builtin_amdgcn_tanhf(x);        // v_tanh_f32 (gfx1250 trans op)
#else
    float e = __expf(2.0f * x);              // v_exp_f32 path
    return 1.0f - 2.0f / (e + 1.0f);
#endif
}

// ---------------------------------------------------------------------------
// Stage 1: fp32 WMMA projections.  out = X[rows,512] @ W[256,512]^T + b
//   z=0: qp  = q @ Wq.T + bq    (row-major [N,ATTN])
//   z=1: kpT = (k @ Wk.T + bk)^T stored [ATTN, M] (coalesced stage-2 reads;
//              per-lane contiguous rows -> two global_store_b128 per lane)
//   z=2: vp  = v @ Wv.T + bv    (row-major [M,ATTN])
// One wave per 16x16 output tile; K-loop of ENC/4 = 128 V_WMMA_F32_16X16X4_F32.
// ---------------------------------------------------------------------------
__global__ __launch_bounds__(128)
void proj_wmma_f32(const float* __restrict__ q, const float* __restrict__ k,
                   const float* __restrict__ v,
                   const float* __restrict__ Wq, const float* __restrict__ bq,
                   const float* __restrict__ Wk, const float* __restrict__ bk,
                   const float* __restrict__ Wv, const float* __restrict__ bv,
                   float* __restrict__ qp, float* __restrict__ kpT,
                   float* __restrict__ vp)
{
    const int z = blockIdx.z;
    const float* X = (z == 0) ? q  : (z == 1) ? k  : v;
    const float* W = (z == 0) ? Wq : (z == 1) ? Wk : Wv;
    const float* B = (z == 0) ? bq : (z == 1) ? bk : bv;
    float*       O = (z == 0) ? qp : (z == 1) ? kpT : vp;

    const int wave   = threadIdx.x >> 5;                  // 0..3
    const int lane   = threadIdx.x & 31;
    const int half   = lane >> 4;                         // 0/1: K-pair select
    const int l16    = lane & 15;
    const int tile_m = blockIdx.y * 16;                   // row tile (N/M dim)
    const int tile_a = (blockIdx.x * 4 + wave) * 16;      // col tile (ATTN dim)

    // A fragment: X[tile_m + l16, kk + half*2 + {0,1}]   (16x4, MxK)
    // B fragment: W[tile_a + l16, kk + half*2 + {0,1}]   (4x16, KxN)
    const float* arow = X + (size_t)(tile_m + l16) * ENC_ + half * 2;
    const float* brow = W + (size_t)(tile_a + l16) * ENC_ + half * 2;

    v8f acc = {};
#pragma unroll 8
    for (int kk = 0; kk < ENC_; kk += 4) {
        v2f a = *(const v2f*)(arow + kk);
        v2f b = *(const v2f*)(brow + kk);
        acc = __builtin_amdgcn_wmma_f32_16x16x4_f32(
            /*neg_a=*/false, a, /*neg_b=*/false, b,
            /*c_mod=*/(short)0, acc, /*reuse_a=*/false, /*reuse_b=*/false);
    }

    // C/D layout: VGPR r, lane L -> M = r + 8*(L>>4), N = L&15
    const int   ncol = tile_a + l16;
    const float bias = B[ncol];

    if (z == 1) {
        // transposed output: lane's 8 elements are contiguous along M
        float* col = O + (size_t)ncol * M_ + tile_m + 8 * half;
        float4 lo = make_float4(acc[0] + bias, acc[1] + bias,
                                acc[2] + bias, acc[3] + bias);
        float4 hi = make_float4(acc[4] + bias, acc[5] + bias,
                                acc[6] + bias, acc[7] + bias);
        *(float4*)(col + 0) = lo;                         // global_store_b128
        *(float4*)(col + 4) = hi;                         // global_store_b128
    } else {
#pragma unroll
        for (int r = 0; r < 8; ++r) {
            const int mrow = tile_m + r + 8 * half;
            O[(size_t)mrow * ATTN_ + ncol] = acc[r] + bias;
        }
    }
}

// ---------------------------------------------------------------------------
// Stage 2: fused additive-attention scores + softmax + context, one block per n.
//   scores[m] = sum_a Ww[a]*tanh(qp[n,a] + kp[m,a]) + bw ; softmax over m
//   context[n,a] = sum_m w[m] * vp[m,a]
// 256 threads = 8 waves. Trans-unit bound (1024*256 tanh per block).
// qp row + Ww staged to LDS via gfx1250 async-to-LDS DMA when available.
// ---------------------------------------------------------------------------
__global__ __launch_bounds__(256)
void attn_fused(const float* __restrict__ qp, const float* __restrict__ kpT,
                const float* __restrict__ vp, const float* __restrict__ Ww,
                const float* __restrict__ bw, const unsigned char* __restrict__ mask,
                float* __restrict__ out)
{
    __shared__ float q_s[ATTN_];
    __shared__ float w_s[ATTN_];
    __shared__ float wt[M_];
    __shared__ float red[8];

    const int t = threadIdx.x;
    const int n = blockIdx.x;

#if HAS_ASYNC_LDS
    __builtin_amdgcn_global_load_async_to_lds_b32(
        (g_i32*)(qp + (size_t)n * ATTN_ + t),
        (s_i32*)&q_s[t], 0, 0);
    __builtin_amdgcn_global_load_async_to_lds_b32(
        (g_i32*)(Ww + t),
        (s_i32*)&w_s[t], 0, 0);
    __builtin_amdgcn_s_wait_asynccnt(0);
#else
    q_s[t] = qp[(size_t)n * ATTN_ + t];
    w_s[t] = Ww[t];
#endif
    __syncthreads();

    const float bias_w = bw[0];

    // ---- scores: each thread owns 4 values of m; 2 accumulators for ILP ----
    float sc[4];
#pragma unroll
    for (int c = 0; c < 4; ++c) {
        const int m = c * 256 + t;
        const float* kcol = kpT + m;                      // stride M_ down a-dim
        float s0 = 0.0f, s1 = 0.0f;
#pragma unroll 4
        for (int a = 0; a < ATTN_; a += 2) {
            s0 += w_s[a]     * fast_tanh(q_s[a]     + kcol[(size_t)a * M_]);
            s1 += w_s[a + 1] * fast_tanh(q_s[a + 1] + kcol[(size_t)(a + 1) * M_]);
        }
        float s = s0 + s1 + bias_w;
        if (!mask[(size_t)n * M_ + m]) s = -1.0e6f;
        sc[c] = s;
    }

    // ---- block max ----
    float mx = fmaxf(fmaxf(sc[0], sc[1]), fmaxf(sc[2], sc[3]));
#pragma unroll
    for (int off = 16; off > 0; off >>= 1)
        mx = fmaxf(mx, __shfl_xor(mx, off, 32));
    if ((t & 31) == 0) red[t >> 5] = mx;
    __syncthreads();
    mx = red[0];
#pragma unroll
    for (int i = 1; i < 8; ++i) mx = fmaxf(mx, red[i]);

    // ---- exp + block sum ----
    float e[4];
    float ls = 0.0f;
#pragma unroll
    for (int c = 0; c < 4; ++c) { e[c] = __expf(sc[c] - mx); ls += e[c]; }
#pragma unroll
    for (int off = 16; off > 0; off >>= 1)
        ls += __shfl_xor(ls, off, 32);
    __syncthreads();                                      // red re-use barrier
    if ((t & 31) == 0) red[t >> 5] = ls;
    __syncthreads();
    float tot = 0.0f;
#pragma unroll
    for (int i = 0; i < 8; ++i) tot += red[i];
    const float inv = 1.0f / tot;

#pragma unroll
    for (int c = 0; c < 4; ++c) wt[c * 256 + t] = e[c] * inv;
    __syncthreads();

    // ---- context GEMV: thread t owns channel a = t; 2 accumulators ----
    float a0 = 0.0f, a1 = 0.0f;
#pragma unroll 4
    for (int m = 0; m < M_; m += 2) {
        a0 += wt[m]     * vp[(size_t)m * ATTN_ + t];
        a1 += wt[m + 1] * vp[(size_t)(m + 1) * ATTN_ + t];
    }
    out[(size_t)n * ATTN_ + t] = a0 + a1;
}

// ---------------------------------------------------------------------------
extern "C" void kernel_launch(void* const* d_in, const int* in_sizes, int n_in,
                              void* d_out, int out_size, void* d_ws, size_t ws_size,
                              hipStream_t stream)
{
    const float*         q    = (const float*)d_in[0];
    const float*         k    = (const float*)d_in[1];
    const float*         v    = (const float*)d_in[2];
    const unsigned char* mask = (const unsigned char*)d_in[3];
    const float*         Wq   = (const float*)d_in[4];
    const float*         bq   = (const float*)d_in[5];
    const float*         Wk   = (const float*)d_in[6];
    const float*         bk   = (const float*)d_in[7];
    const float*         Wv   = (const float*)d_in[8];
    const float*         bv   = (const float*)d_in[9];
    const float*         Ww   = (const float*)d_in[10];
    const float*         bw   = (const float*)d_in[11];
    float*               out  = (float*)d_out;

    float* qp  = (float*)d_ws;                 // [N, ATTN]   1 MB
    float* kpT = qp  + (size_t)N_ * ATTN_;     // [ATTN, M]   1 MB (transposed)
    float* vp  = kpT + (size_t)ATTN_ * M_;     // [M, ATTN]   1 MB

    dim3 gproj(ATTN_ / 64, N_ / 16, 3);        // 4 x 64 x 3 blocks, 4 waves each
    proj_wmma_f32<<<gproj, 128, 0, stream>>>(q, k, v, Wq, bq, Wk, bk, Wv, bv,
                                             qp, kpT, vp);

    attn_fused<<<N_, 256, 0, stream>>>(qp, kpT, vp, Ww, bw, mask, out);
}